// TSM_75196287418795
// MI455X (gfx1250) — compile-verified
//
#include <hip/hip_runtime.h>
#include <cstdint>

// TSM temporal shift for x:(128, 96, 56, 56) f32, N_FRAME=8, FOLD_DIV=3 (fold=32).
// out[b,c,:,:] = in[b+dt, c, :, :]; dt = +1 (c<32, zero at t==7),
//                                   dt = -1 (32<=c<64, zero at t==0),
//                                   dt =  0 (c>=64).
// 0 FLOPs, ~308 MB HBM traffic -> ~13 us floor at 23.3 TB/s. One block per
// (b,c) plane: shift logic is block-uniform SALU, no divergence. Bulk copy is
// the CDNA5 async data mover (ASYNCcnt) in GVS addressing: SGPR-pair plane
// base + one per-lane 32-bit offset VGPR, with immediate offset: covering the
// 784-vec4 plane (784 = 3*256 + 16). INST_OFFSET advances BOTH the LDS and
// global addresses (CDNA5 ISA), so one register pair serves all
// issues.

#define TSM_C         96
#define TSM_FOLD      32
#define TSM_NF        8
#define TSM_HW        3136                     // 56*56 floats per plane
#define VEC_PER_PLANE (TSM_HW / 4)             // 784 float4 per plane
#define FRAME_VEC     (TSM_C * VEC_PER_PLANE)  // 75264 float4 per frame

__global__ __launch_bounds__(256) void tsm_shift_kernel(const float* __restrict__ in,
                                                        float* __restrict__ out) {
    __shared__ alignas(16) unsigned char lds_buf[VEC_PER_PLANE * 16]; // 12544 B

    const int tid   = threadIdx.x;
    const int plane = blockIdx.x;          // b*96 + c (block-uniform)
    const int c     = plane % TSM_C;       // uniform -> SALU
    const int b     = plane / TSM_C;
    const int t     = b & (TSM_NF - 1);

    int  delta;
    bool zero;
    if (c < TSM_FOLD)            { delta = +1; zero = (t == TSM_NF - 1); }
    else if (c < 2 * TSM_FOLD)   { delta = -1; zero = (t == 0);          }
    else                         { delta =  0; zero = false;             }

    const int base = plane * VEC_PER_PLANE;
    float4* __restrict__ dstp = (float4*)out + base;

    if (zero) {
        // Uniform branch: whole plane zero-filled with global_store_b128.
        const float4 z = make_float4(0.f, 0.f, 0.f, 0.f);
        dstp[tid]       = z;
        dstp[tid + 256] = z;
        dstp[tid + 512] = z;
        if (tid < VEC_PER_PLANE - 768) dstp[tid + 768] = z;
        return;
    }

    const float4* __restrict__ srcp = (const float4*)in + (base + delta * FRAME_VEC);

    // Per-lane 32-bit byte offset; same value offsets the LDS slot and the
    // global address (GVS mode: mem = SADDR + VADDR + INST_OFFSET, and the
    // LDS side gets VDST + INST_OFFSET).
    const unsigned int voff    = (unsigned int)tid * 16u;
    const unsigned int lds_off = (unsigned int)(uintptr_t)lds_buf + voff;

    // Phase 1: async loads HBM -> LDS, 3-4 in flight per lane.
    asm volatile("global_load_async_to_lds_b128 %0, %1, %2"
                 :: "v"(lds_off), "v"(voff), "s"(srcp) : "memory");
    asm volatile("global_load_async_to_lds_b128 %0, %1, %2 offset:4096"
                 :: "v"(lds_off), "v"(voff), "s"(srcp) : "memory");
    asm volatile("global_load_async_to_lds_b128 %0, %1, %2 offset:8192"
                 :: "v"(lds_off), "v"(voff), "s"(srcp) : "memory");
    if (tid < VEC_PER_PLANE - 768) {
        asm volatile("global_load_async_to_lds_b128 %0, %1, %2 offset:12288"
                     :: "v"(lds_off), "v"(voff), "s"(srcp) : "memory");
    }

    // Drain this wave's async loads; each lane stores only its own slots,
    // so no workgroup barrier is needed.
    asm volatile("s_wait_asynccnt 0" ::: "memory");

    // Phase 2: pipelined async stores LDS -> HBM; S_ENDPGM's implicit
    // wait-idle drains ASYNCcnt at wave exit.
    asm volatile("global_store_async_from_lds_b128 %0, %1, %2"
                 :: "v"(voff), "v"(lds_off), "s"(dstp) : "memory");
    asm volatile("global_store_async_from_lds_b128 %0, %1, %2 offset:4096"
                 :: "v"(voff), "v"(lds_off), "s"(dstp) : "memory");
    asm volatile("global_store_async_from_lds_b128 %0, %1, %2 offset:8192"
                 :: "v"(voff), "v"(lds_off), "s"(dstp) : "memory");
    if (tid < VEC_PER_PLANE - 768) {
        asm volatile("global_store_async_from_lds_b128 %0, %1, %2 offset:12288"
                     :: "v"(voff), "v"(lds_off), "s"(dstp) : "memory");
    }
}

extern "C" void kernel_launch(void* const* d_in, const int* in_sizes, int n_in,
                              void* d_out, int out_size, void* d_ws, size_t ws_size,
                              hipStream_t stream) {
    const float* x = (const float*)d_in[0];
    float* out     = (float*)d_out;

    const int planes = out_size / TSM_HW;   // 128*96 = 12288 blocks
    tsm_shift_kernel<<<planes, 256, 0, stream>>>(x, out);
}